// EMAVectorQuantizer_12979391169318
// MI455X (gfx1250) — compile-verified
//
#include <hip/hip_runtime.h>
#include <hip/hip_bf16.h>

typedef __attribute__((ext_vector_type(2))) float v2f;
typedef __attribute__((ext_vector_type(8))) float v8f;

#define N_ROWS   65536
#define K_CODES  2048
#define EDIM     256
#define CHUNK    32          // codes staged in LDS per iteration
#define LDS_STR  260         // padded row stride (floats) -> conflict-free b64 LDS reads
#define ROWS_PER_BLOCK 128   // 8 waves * 16 rows

// ---------------------------------------------------------------------------
// Kernel: zero workspace (loss accumulator + n_cnt + dw)
// ---------------------------------------------------------------------------
__global__ void vq_zero_kernel(float* __restrict__ p, int n) {
  int i = blockIdx.x * blockDim.x + threadIdx.x;
  if (i < n) p[i] = 0.0f;
}

// ---------------------------------------------------------------------------
// Kernel: e_sq[c] = sum_d emb[c][d]^2   (one block per code)
// ---------------------------------------------------------------------------
__global__ __launch_bounds__(256) void vq_esq_kernel(const float* __restrict__ emb,
                                                     float* __restrict__ e_sq) {
  __shared__ float red[256];
  const int c = blockIdx.x, d = threadIdx.x;
  float e = emb[(size_t)c * EDIM + d];
  red[d] = e * e;
  __syncthreads();
  for (int s = 128; s > 0; s >>= 1) {
    if (d < s) red[d] += red[d + s];
    __syncthreads();
  }
  if (d == 0) e_sq[c] = red[0];
}

// ---------------------------------------------------------------------------
// Kernel: fused distance GEMM + argmin via V_WMMA_F32_16X16X4_F32.
// Each wave: 16 rows of z_e (A panel resident in 128 VGPRs), streams all
// 2048 codes through LDS, tracks per-row running min of (||e||^2 - 2 z.e).
// ---------------------------------------------------------------------------
__global__ __launch_bounds__(256) void vq_argmin_kernel(
    const float* __restrict__ z_e, const float* __restrict__ emb,
    const float* __restrict__ e_sq, int* __restrict__ idx_out,
    float* __restrict__ idx_f_out) {
  __shared__ float lds_e[CHUNK * LDS_STR];
  __shared__ float lds_q[CHUNK];

  const int tid  = threadIdx.x;
  const int lane = tid & 31;
  const int wave = tid >> 5;
  const int m16  = lane & 15;   // M (for A) / N (for B,C) within tile
  const int h    = lane >> 4;   // half-wave selects K pair
  const int row_base = blockIdx.x * ROWS_PER_BLOCK + wave * 16;

  // Preload A panel: af[s] = { z[m16][4s+2h], z[m16][4s+2h+1] }  (ISA 16x4 f32 layout)
  v2f af[64];
  const float* arow = z_e + (size_t)(row_base + m16) * EDIM + 2 * h;
#pragma unroll
  for (int s = 0; s < 64; ++s)
    af[s] = *reinterpret_cast<const v2f*>(arow + 4 * s);

  float vmin[8];
  int   imin[8];
#pragma unroll
  for (int r = 0; r < 8; ++r) { vmin[r] = 3.4028235e38f; imin[r] = 0; }

  for (int c0 = 0; c0 < K_CODES; c0 += CHUNK) {
    __syncthreads();
    // cooperative stage of CHUNK codes into padded LDS
    for (int i = tid; i < CHUNK * EDIM; i += 256) {
      int rr = i >> 8, cc = i & 255;
      lds_e[rr * LDS_STR + cc] = emb[(size_t)(c0 + rr) * EDIM + cc];
    }
    if (tid < CHUNK) lds_q[tid] = e_sq[c0 + tid];
    __syncthreads();

#pragma unroll
    for (int t = 0; t < CHUNK / 16; ++t) {
      v8f acc = {};
      const float* brow = &lds_e[(t * 16 + m16) * LDS_STR + 2 * h];
#pragma unroll
      for (int s = 0; s < 64; ++s) {
        v2f b = *reinterpret_cast<const v2f*>(brow + 4 * s);
        acc = __builtin_amdgcn_wmma_f32_16x16x4_f32(
            false, af[s], false, b, (short)0, acc, false, false);
      }
      const float q = lds_q[t * 16 + m16];
      const int   n = c0 + t * 16 + m16;
#pragma unroll
      for (int r = 0; r < 8; ++r) {
        float score = q - 2.0f * acc[r];      // ||e||^2 - 2 <z,e>
        if (score < vmin[r]) { vmin[r] = score; imin[r] = n; }
      }
    }
  }

  // reduce across the 16 lanes of each half (C layout: lane half h -> M = r + 8h)
#pragma unroll
  for (int r = 0; r < 8; ++r) {
#pragma unroll
    for (int m = 1; m < 16; m <<= 1) {
      float ov = __shfl_xor(vmin[r], m, 16);
      int   oi = __shfl_xor(imin[r], m, 16);
      if (ov < vmin[r] || (ov == vmin[r] && oi < imin[r])) {
        vmin[r] = ov; imin[r] = oi;
      }
    }
  }
  if (m16 == 0) {
#pragma unroll
    for (int r = 0; r < 8; ++r) {
      int row = row_base + r + 8 * h;
      idx_out[row]   = imin[r];
      idx_f_out[row] = (float)imin[r];
    }
  }
}

// ---------------------------------------------------------------------------
// Kernel: gather z_q, write straight-through output, accumulate stats.
// One block (256 threads) per row; one thread per dim.
// ---------------------------------------------------------------------------
__global__ __launch_bounds__(256) void vq_scatter_kernel(
    const float* __restrict__ z_e, const float* __restrict__ emb,
    const int* __restrict__ idx, float* __restrict__ zq_out,
    float* __restrict__ dw, int* __restrict__ n_cnt, double* __restrict__ loss) {
  __shared__ float red[256];
  const int row = blockIdx.x;
  const int d   = threadIdx.x;
  const int c   = idx[row];
  const float z = z_e[(size_t)row * EDIM + d];
  const float e = emb[(size_t)c * EDIM + d];
  zq_out[(size_t)row * EDIM + d] = z + (e - z);   // z_e + stop_grad(z_q - z_e)
  const float diff = e - z;
  red[d] = diff * diff;
  atomicAdd(&dw[(size_t)c * EDIM + d], z);
  __syncthreads();
  for (int s = 128; s > 0; s >>= 1) {
    if (d < s) red[d] += red[d + s];
    __syncthreads();
  }
  if (d == 0) {
    atomicAdd(loss, (double)red[0]);
    atomicAdd(&n_cnt[c], 1);   // integer atomic: exact + deterministic
  }
}

// ---------------------------------------------------------------------------
// Kernel: EMA codebook update (one block per code)
// ---------------------------------------------------------------------------
__global__ __launch_bounds__(256) void vq_ema_kernel(
    const float* __restrict__ ema_w, const float* __restrict__ cs,
    const float* __restrict__ dw, const int* __restrict__ n_cnt,
    float* __restrict__ new_emb, float* __restrict__ new_cs,
    float* __restrict__ new_ema_w) {
  const int c = blockIdx.x, d = threadIdx.x;
  const float nj  = (float)n_cnt[c];
  const float ncs = 0.99f * cs[c] + 0.01f * nj;
  const float nw  = 0.99f * ema_w[(size_t)c * EDIM + d] + 0.01f * dw[(size_t)c * EDIM + d];
  new_ema_w[(size_t)c * EDIM + d] = nw;
  const float cl = ncs < 1e-5f ? 1e-5f : ncs;
  new_emb[(size_t)c * EDIM + d] = nw / cl;
  if (d == 0) new_cs[c] = ncs;
}

// ---------------------------------------------------------------------------
// Kernel: perplexity + finalize commitment loss (single block)
// ---------------------------------------------------------------------------
__global__ __launch_bounds__(256) void vq_final_kernel(
    const int* __restrict__ n_cnt, const double* __restrict__ loss,
    float* __restrict__ out_loss, float* __restrict__ out_perp) {
  __shared__ float red[256];
  float s = 0.0f;
  for (int c = threadIdx.x; c < K_CODES; c += 256) {
    float p = (float)n_cnt[c] * (1.0f / (float)N_ROWS);
    s += p * __logf(p + 1e-10f);
  }
  red[threadIdx.x] = s;
  __syncthreads();
  for (int t = 128; t > 0; t >>= 1) {
    if (threadIdx.x < t) red[threadIdx.x] += red[threadIdx.x + t];
    __syncthreads();
  }
  if (threadIdx.x == 0) {
    *out_perp = __expf(-red[0]);
    *out_loss = 0.1f * (float)(*loss / ((double)N_ROWS * (double)EDIM));
  }
}

// ---------------------------------------------------------------------------
// Host launcher
// ---------------------------------------------------------------------------
extern "C" void kernel_launch(void* const* d_in, const int* in_sizes, int n_in,
                              void* d_out, int out_size, void* d_ws, size_t ws_size,
                              hipStream_t stream) {
  const float* z_e   = (const float*)d_in[0];
  const float* emb   = (const float*)d_in[1];
  const float* cs    = (const float*)d_in[2];
  const float* ema_w = (const float*)d_in[3];

  float* out = (float*)d_out;
  // output tuple, flattened in return order
  float* o_zq   = out;                                  // [N, D]
  float* o_idx  = o_zq  + (size_t)N_ROWS * EDIM;        // [N]
  float* o_loss = o_idx + N_ROWS;                       // scalar
  float* o_perp = o_loss + 1;                           // scalar
  float* o_nemb = o_perp + 1;                           // [K, D]
  float* o_ncs  = o_nemb + (size_t)K_CODES * EDIM;      // [K]
  float* o_new_ema = o_ncs + K_CODES;                   // [K, D]

  // workspace layout
  float*  wsf    = (float*)d_ws;
  double* ws_loss = (double*)d_ws;               // wsf[0..1]
  float*  ws_esq  = wsf + 2;                     // [K]
  int*    ws_cnt  = (int*)(wsf + 2 + K_CODES);   // [K]
  float*  ws_dw   = wsf + 2 + 2 * K_CODES;       // [K, D]
  int*    ws_idx  = (int*)(wsf + 2 + 2 * K_CODES + (size_t)K_CODES * EDIM); // [N]

  // zero loss + counts + dw (e_sq and idx are fully overwritten each call)
  const int nzero = 2 + 2 * K_CODES + K_CODES * EDIM;
  vq_zero_kernel<<<(nzero + 255) / 256, 256, 0, stream>>>(wsf, nzero);

  vq_esq_kernel<<<K_CODES, 256, 0, stream>>>(emb, ws_esq);

  vq_argmin_kernel<<<N_ROWS / ROWS_PER_BLOCK, 256, 0, stream>>>(
      z_e, emb, ws_esq, ws_idx, o_idx);

  vq_scatter_kernel<<<N_ROWS, 256, 0, stream>>>(
      z_e, emb, ws_idx, o_zq, ws_dw, ws_cnt, ws_loss);

  vq_ema_kernel<<<K_CODES, 256, 0, stream>>>(
      ema_w, cs, ws_dw, ws_cnt, o_nemb, o_ncs, o_new_ema);

  vq_final_kernel<<<1, 256, 0, stream>>>(ws_cnt, ws_loss, o_loss, o_perp);
}